// InvRT_45406394253466
// MI455X (gfx1250) — compile-verified
//
#include <hip/hip_runtime.h>
#include <hip/hip_bf16.h>

typedef __attribute__((ext_vector_type(4))) float v4f;

#ifndef __has_builtin
#define __has_builtin(x) 0
#endif

// gfx1250 adds a hardware tanh (V_TANH_F32). Prefer it; fall back to libm.
__device__ __forceinline__ float fast_tanh(float x) {
#if __has_builtin(__builtin_amdgcn_tanhf)
  return __builtin_amdgcn_tanhf(x);
#elif __has_builtin(__builtin_amdgcn_tanh_f32)
  return __builtin_amdgcn_tanh_f32(x);
#else
  return tanhf(x);
#endif
}

namespace {
constexpr int kM = 8;
constexpr int kN = 64;
constexpr int kB = 512;
constexpr int kL = 128;
constexpr long kNB = (long)kN * kB;  // rows of length L per m-slice: 32768
}  // namespace

// out[m,n,b,l] = -(e0 + e1 * tanh((z - e2) * e3)), e* = eta_fault[Mask[m,l]]
// Thread (tx, ty): tx in [0,32) owns l = 4*tx .. 4*tx+3 (one float4 along the
// contiguous axis); eta coefficients for those 4 l's are loaded once into
// registers. Then grid-stride over the (n,b) rows of this block's m-slice,
// doing one nontemporal b128 load + 4 tanh/fma quads + one nontemporal b128
// store per row. Purely bandwidth-bound: ~256 MiB / 23.3 TB/s ≈ 11 us floor.
__global__ __launch_bounds__(256) void eta_tanh_stream(
    const float* __restrict__ z, const float* __restrict__ eta,
    const long long* __restrict__ mask, float* __restrict__ out) {
  const int tx = threadIdx.x;  // 0..31 -> which quad of L
  const int m = blockIdx.y;    // 0..7
  const int l0 = tx << 2;

  // One-time gather of the 4 eta rows this thread needs (table is 304 B,
  // stays resident in cache; cost is amortized over 4 full rows of work).
  float e0[4], e1[4], e2[4], e3[4];
#pragma unroll
  for (int j = 0; j < 4; ++j) {
    const int row = (int)mask[(size_t)m * kL + l0 + j];  // 0..18
    const v4f er = *(const v4f*)(eta + row * 4);
    e0[j] = er.x;
    e1[j] = er.y;
    e2[j] = er.z;
    e3[j] = er.w;
  }

  const size_t rstride = (size_t)gridDim.x * blockDim.y;
  for (size_t r = (size_t)blockIdx.x * blockDim.y + threadIdx.y;
       r < (size_t)kNB; r += rstride) {
    const size_t base = ((size_t)m * kNB + r) * kL + l0;
    const v4f zv = __builtin_nontemporal_load((const v4f*)(z + base));
    v4f ov;
    ov.x = -(e0[0] + e1[0] * fast_tanh((zv.x - e2[0]) * e3[0]));
    ov.y = -(e0[1] + e1[1] * fast_tanh((zv.y - e2[1]) * e3[1]));
    ov.z = -(e0[2] + e1[2] * fast_tanh((zv.z - e2[2]) * e3[2]));
    ov.w = -(e0[3] + e1[3] * fast_tanh((zv.w - e2[3]) * e3[3]));
    __builtin_nontemporal_store(ov, (v4f*)(out + base));
  }
}

extern "C" void kernel_launch(void* const* d_in, const int* in_sizes, int n_in,
                              void* d_out, int out_size, void* d_ws,
                              size_t ws_size, hipStream_t stream) {
  (void)in_sizes;
  (void)n_in;
  (void)out_size;
  (void)d_ws;
  (void)ws_size;
  const float* z = (const float*)d_in[0];           // (8,64,512,128) f32
  const float* eta = (const float*)d_in[1];         // (19,4) f32
  const long long* mask = (const long long*)d_in[2];  // (8,128) i64
  float* out = (float*)d_out;                       // (8,64,512,128) f32

  dim3 block(32, 8);    // 256 threads = 8 waves (wave32)
  dim3 grid(1024, kM);  // 8192 rows in flight per m-sweep -> 4 rows/thread
  hipLaunchKernelGGL(eta_tanh_stream, grid, block, 0, stream, z, eta, mask,
                     out);
}